// MarginDevianceLoss_83794811945706
// MI455X (gfx1250) — compile-verified
//
#include <hip/hip_runtime.h>
#include <math.h>

// Problem constants from the reference
#define N_TOT 4096   // batch
#define DIM   1024   // embedding dim
#define KCLS  8      // instances per class
#define PCNT  7.0f      // K-1 positives per row
#define NCNT  4088.0f   // N-K negatives per row

typedef float v2f __attribute__((ext_vector_type(2)));
typedef float v8f __attribute__((ext_vector_type(8)));
typedef unsigned int u32x4 __attribute__((ext_vector_type(4)));
typedef int i32x4 __attribute__((ext_vector_type(4)));
typedef int i32x8 __attribute__((ext_vector_type(8)));

// A-strip in LDS: 16 rows x 1024 f32, TDM pads 1 dword per 256 dwords
// -> effective row stride 1028 dwords (1028 % 64 == 4: conflict-free ds_load_b64)
#define A_TILE_DW (16 * 1024 + 16 * 4)

__device__ __forceinline__ float softplus_f(float x) {
  // stable log(1+exp(x))
  return fmaxf(x, 0.0f) + log1pf(__expf(-fabsf(x)));
}

// Stage one 16x1024 f32 strip (row-major, stride 1024) from global into LDS
// via the Tensor Data Mover. 2D tile: tile_dim0=1024 elems, tile_dim1=16 rows,
// data_size=4B, pad_enable, pad_interval=256 dw, pad_amount=1 dw.
__device__ __forceinline__ void tdm_load_strip(const float* gsrc, unsigned lds_byte_off) {
  unsigned long long ga = (unsigned long long)gsrc;
  u32x4 g0;
  g0.x = 1u;                                   // count=1, user descriptor
  g0.y = lds_byte_off;                         // lds_addr
  g0.z = (unsigned)ga;                         // global_addr[31:0]
  g0.w = (unsigned)(ga >> 32) | (2u << 30);    // global_addr[56:32] | type=2
  i32x8 g1;
  g1[0] = (2 << 16) | (1 << 20) | (7 << 22);   // data_size=4B, pad_en, interval=256dw, amount=1dw
  g1[1] = (1024 << 16);                        // tensor_dim0 = 1024 (low 16 @ bits 63:48)
  g1[2] = (4096 << 16);                        // tensor_dim0 hi=0, tensor_dim1 = 4096 (low 16)
  g1[3] = (1024 << 16);                        // tensor_dim1 hi=0, tile_dim0 = 1024
  g1[4] = 16;                                  // tile_dim1 = 16, tile_dim2 = 0
  g1[5] = 1024;                                // tensor_dim0_stride[31:0] = 1024 elems
  g1[6] = 0;                                   // stride0 hi, stride1 lo
  g1[7] = 0;                                   // stride1 hi
  i32x4 gz = {0, 0, 0, 0};
#if __clang_major__ >= 23
  i32x8 gz8 = {0, 0, 0, 0, 0, 0, 0, 0};
  __builtin_amdgcn_tensor_load_to_lds(g0, g1, gz, gz, gz8, 0);
#else
  __builtin_amdgcn_tensor_load_to_lds(g0, g1, gz, gz, 0);
#endif
}

// PASS 1: GEMM strip + per-row masked mean/std -> inter/thresh + pos/neg sums
// PASS 2: GEMM strip (recompute) + softplus losses with hard-negative filter
template <int PASS>
__global__ void md_pass_kernel(const float* __restrict__ X,
                               const int* __restrict__ tgt,
                               float* __restrict__ ws_inter,
                               float* __restrict__ ws_thresh,
                               float* __restrict__ acc) {
  extern __shared__ float smem[];
  float* ldsA = smem;                       // A_TILE_DW dwords
  float* rowA = smem + A_TILE_DW;           // 16: pos_sum   / pos_loss
  float* rowB = rowA + 16;                  // 16: pos_sumsq / neg_loss
  float* rowC = rowB + 16;                  // 16: neg_sum   / keep_cnt
  float* rowD = rowC + 16;                  // 16: neg_sumsq / (unused)
  int* trowS = (int*)(rowD + 16);           // 16: row targets
  float* interS = (float*)(trowS + 16);     // 16
  float* thrS = interS + 16;                // 16

  const int strip = blockIdx.x;             // 0..255, 16 rows each
  const int tid = threadIdx.x;
  const int wave = tid >> 5;                // 8 waves of 32 (wave32)
  const int lane = tid & 31;
  const int lo = lane & 15;
  const int hi = lane >> 4;
  const int rowbase = strip * 16;

  if (tid < 16) {
    rowA[tid] = 0.0f; rowB[tid] = 0.0f; rowC[tid] = 0.0f; rowD[tid] = 0.0f;
    trowS[tid] = tgt[rowbase + tid];
    if (PASS == 2) {
      interS[tid] = ws_inter[rowbase + tid];
      thrS[tid] = ws_thresh[rowbase + tid];
    }
  }

  // Wave 0 stages the A strip via TDM; TENSORcnt fence before the barrier.
  if (wave == 0) {
    tdm_load_strip(X + (size_t)rowbase * DIM, (unsigned)(size_t)ldsA);
    __builtin_amdgcn_s_wait_tensorcnt(0);
  }
  __syncthreads();

  // Hoist per-lane row metadata (fixed across the j-tile loop).
  // C/D layout: VGPR r -> row (r + 8*hi), col (jt*16 + lo).
  int ti_r[8];
  float inter_r[8], thr_r[8];
#pragma unroll
  for (int r = 0; r < 8; ++r) {
    const int iloc = r + 8 * hi;
    ti_r[r] = trowS[iloc];
    if (PASS == 2) {
      inter_r[r] = interS[iloc];
      thr_r[r] = thrS[iloc];
    } else {
      inter_r[r] = 0.0f;
      thr_r[r] = 0.0f;
    }
  }

  float accA[8], accB[8], accC[8], accD[8];
#pragma unroll
  for (int r = 0; r < 8; ++r) { accA[r] = 0.f; accB[r] = 0.f; accC[r] = 0.f; accD[r] = 0.f; }

  // A fragment base in the padded LDS image:
  //   off_dw(lo*1024 + 2*hi + kc) = lo*1028 + 2*hi + kc + (kc>>8)
  // so within a 256-dword K-segment all DS offsets are immediates.
  const float* abase = ldsA + (unsigned)(lo * 1028 + 2 * hi);

  // Each wave owns 32 column tiles, processed as 8 groups of 4 tiles
  // (4 independent C accumulators -> 4x A-fragment reuse, 4 independent
  //  WMMA dependence chains).
  for (int g = 0; g < 8; ++g) {
    const int jt0 = wave * 32 + g * 4;

    int jcol[4], tj[4];
    const float* bptr[4];
#pragma unroll
    for (int q = 0; q < 4; ++q) {
      jcol[q] = (jt0 + q) * 16 + lo;
      tj[q] = tgt[jcol[q]];
      bptr[q] = X + (size_t)jcol[q] * DIM + 2 * hi;
      // Prefetch next group's B rows (one byte/lane pulls whole cachelines).
      if (g < 7) __builtin_prefetch(bptr[q] + DIM * 64, 0, 1);
    }

    // Opaque zero: redefined every group iteration so the A-fragment LDS
    // loads are not provably loop-invariant -> no LICM hoist + scratch spill.
    unsigned tz = 0;
    asm volatile("" : "+v"(tz));
    const float* abase_g = abase + tz;

    v8f c[4];
#pragma unroll
    for (int q = 0; q < 4; ++q) c[q] = (v8f){0.f, 0.f, 0.f, 0.f, 0.f, 0.f, 0.f, 0.f};

#pragma unroll
    for (int seg = 0; seg < 4; ++seg) {
      const int ks = seg * 256;
      const float* aseg = abase_g + ks + seg;  // +seg = pad dwords before segment
      const float* bs0 = bptr[0] + ks;
      const float* bs1 = bptr[1] + ks;
      const float* bs2 = bptr[2] + ks;
      const float* bs3 = bptr[3] + ks;

      // Depth-2 software pipeline: loads for k+4 issued before the WMMAs for k.
      v2f a0 = *(const v2f*)(aseg);
      v2f b0 = *(const v2f*)(bs0);
      v2f b1 = *(const v2f*)(bs1);
      v2f b2 = *(const v2f*)(bs2);
      v2f b3 = *(const v2f*)(bs3);
#pragma unroll 4
      for (int k2 = 0; k2 < 252; k2 += 4) {
        v2f an = *(const v2f*)(aseg + k2 + 4);
        v2f bn0 = *(const v2f*)(bs0 + k2 + 4);
        v2f bn1 = *(const v2f*)(bs1 + k2 + 4);
        v2f bn2 = *(const v2f*)(bs2 + k2 + 4);
        v2f bn3 = *(const v2f*)(bs3 + k2 + 4);
        c[0] = __builtin_amdgcn_wmma_f32_16x16x4_f32(false, a0, false, b0, (short)0,
                                                     c[0], false, false);
        c[1] = __builtin_amdgcn_wmma_f32_16x16x4_f32(false, a0, false, b1, (short)0,
                                                     c[1], false, false);
        c[2] = __builtin_amdgcn_wmma_f32_16x16x4_f32(false, a0, false, b2, (short)0,
                                                     c[2], false, false);
        c[3] = __builtin_amdgcn_wmma_f32_16x16x4_f32(false, a0, false, b3, (short)0,
                                                     c[3], false, false);
        a0 = an; b0 = bn0; b1 = bn1; b2 = bn2; b3 = bn3;
      }
      c[0] = __builtin_amdgcn_wmma_f32_16x16x4_f32(false, a0, false, b0, (short)0,
                                                   c[0], false, false);
      c[1] = __builtin_amdgcn_wmma_f32_16x16x4_f32(false, a0, false, b1, (short)0,
                                                   c[1], false, false);
      c[2] = __builtin_amdgcn_wmma_f32_16x16x4_f32(false, a0, false, b2, (short)0,
                                                   c[2], false, false);
      c[3] = __builtin_amdgcn_wmma_f32_16x16x4_f32(false, a0, false, b3, (short)0,
                                                   c[3], false, false);
    }

    // Masked-statistics epilogue for the 4 tiles of this group.
#pragma unroll
    for (int q = 0; q < 4; ++q) {
#pragma unroll
      for (int r = 0; r < 8; ++r) {
        const int i = rowbase + r + 8 * hi;
        const float v = c[q][r];
        const bool same = (ti_r[r] == tj[q]);
        const bool diag = (i == jcol[q]);
        if (PASS == 1) {
          const float v2 = v * v;
          if (same && !diag) { accA[r] += v; accB[r] += v2; }
          if (!same)         { accC[r] += v; accD[r] += v2; }
        } else {
          const float d = v - inter_r[r];
          if (same && !diag) accA[r] += softplus_f(-2.0f * d);
          if ((!same) && (v > thr_r[r])) {
            accB[r] += softplus_f(50.0f * d);
            accC[r] += 1.0f;
          }
        }
      }
    }
  }

  // Reduce per-lane partials into per-row LDS accumulators.
#pragma unroll
  for (int r = 0; r < 8; ++r) {
    const int iloc = r + 8 * hi;
    atomicAdd(&rowA[iloc], accA[r]);
    atomicAdd(&rowB[iloc], accB[r]);
    atomicAdd(&rowC[iloc], accC[r]);
    if (PASS == 1) atomicAdd(&rowD[iloc], accD[r]);
  }
  __syncthreads();

  if (tid < 16) {
    if (PASS == 1) {
      const float ps = rowA[tid], psq = rowB[tid];
      const float ns = rowC[tid], nsq = rowD[tid];
      const float pmean = ps / PCNT;
      const float pstd = sqrtf(fmaxf(psq / PCNT - pmean * pmean, 0.0f));
      const float nmean = ns / NCNT;
      const float nstd = sqrtf(fmaxf(nsq / NCNT - nmean * nmean, 0.0f));
      ws_inter[rowbase + tid] = (pstd * pmean + nstd * nmean) / (pstd + nstd);
      ws_thresh[rowbase + tid] = pmean - 1.5f * pstd;
      atomicAdd(&acc[2], ps);   // pos_d numerator
      atomicAdd(&acc[3], ns);   // neg_d numerator
    } else {
      const float pl = rowA[tid], nl = rowB[tid], kcnt = rowC[tid];
      const bool valid = (kcnt >= 1.0f);
      const float contrib =
          valid ? (pl / PCNT + 0.04f * nl / fmaxf(kcnt, 1.0f)) : 0.0f;
      atomicAdd(&acc[0], contrib);
      if (!valid) atomicAdd(&acc[1], 1.0f);
    }
  }
}

__global__ void md_init_acc(float* acc) {
  if (threadIdx.x < 4) acc[threadIdx.x] = 0.0f;
}

__global__ void md_finalize(const float* __restrict__ acc, float* __restrict__ out) {
  if (threadIdx.x == 0) {
    out[0] = acc[0] / (float)N_TOT;                 // loss
    out[1] = acc[1] / (float)N_TOT;                 // prec (fraction invalid)
    out[2] = acc[2] / ((float)N_TOT * PCNT);        // pos_d
    out[3] = acc[3] / ((float)N_TOT * NCNT);        // neg_d
  }
}

extern "C" void kernel_launch(void* const* d_in, const int* in_sizes, int n_in,
                              void* d_out, int out_size, void* d_ws, size_t ws_size,
                              hipStream_t stream) {
  (void)in_sizes; (void)n_in; (void)out_size; (void)ws_size;
  const float* X = (const float*)d_in[0];
  const int* tgt = (const int*)d_in[1];
  float* ws = (float*)d_ws;
  float* ws_inter = ws;                 // [4096]
  float* ws_thresh = ws + N_TOT;        // [4096]
  float* acc = ws + 2 * N_TOT;          // [4]
  float* out = (float*)d_out;

  const size_t shmem = (size_t)(A_TILE_DW + 16 * 4 + 16 + 32) * sizeof(float);

  md_init_acc<<<1, 32, 0, stream>>>(acc);
  md_pass_kernel<1><<<N_TOT / 16, 256, shmem, stream>>>(X, tgt, ws_inter, ws_thresh, acc);
  md_pass_kernel<2><<<N_TOT / 16, 256, shmem, stream>>>(X, tgt, ws_inter, ws_thresh, acc);
  md_finalize<<<1, 32, 0, stream>>>(acc, out);
}